// EnsembleGCN_42984032698665
// MI455X (gfx1250) — compile-verified
//
#include <hip/hip_runtime.h>
#include <hip/hip_bf16.h>

#define NN      2048
#define NM1     2047
#define DT      256
#define ET      32
#define EC      128
#define NCLS    5
#define SUPPORT 1536   /* n - query_size */

#define CH   32        /* K-rows of Zs staged per LDS chunk   */
#define ZLD  136       /* LDS row stride (floats): 2*ZLD%64==16 -> conflict-free halves */

typedef float v2f __attribute__((ext_vector_type(2)));
typedef float v8f __attribute__((ext_vector_type(8)));
typedef int   v4i __attribute__((ext_vector_type(4)));

#if __has_builtin(__builtin_amdgcn_global_load_async_to_lds_b128) && \
    __has_builtin(__builtin_amdgcn_s_wait_asynccnt)
#define USE_ASYNC_LDS 1
typedef __attribute__((address_space(1))) v4i* g_v4i_p;   // global (AS1) int4*
typedef __attribute__((address_space(3))) v4i* l_v4i_p;   // LDS (AS3) int4*
#else
#define USE_ASYNC_LDS 0
#endif

static __device__ __forceinline__ v8f wmma_f32_k4(v2f a, v2f b, v8f c) {
  // D = A(16x4 f32) x B(4x16 f32) + C(16x16 f32), full fp32 precision
  return __builtin_amdgcn_wmma_f32_16x16x4_f32(
      /*neg_a=*/false, a, /*neg_b=*/false, b,
      /*c_mod=*/(short)0, c, /*reuse_a=*/false, /*reuse_b=*/false);
}

static __device__ __forceinline__ float lrelu(float x) {
  return x > 0.0f ? x : 0.01f * x;
}

// ---------------------------------------------------------------- init ----
__global__ void k_init(float* deg_t, float* deg_f, float* degw, float* cols) {
  int t = blockIdx.x * blockDim.x + threadIdx.x;
  if (t < NN) { deg_t[t] = 0.0f; deg_f[t] = 0.0f; degw[t] = 0.0f; }
  if (t < 8) cols[t] = 0.0f;
}

// ------------------------------------------------- XW = X[2048,256]@W[256,32]
__global__ void k_xw(const float* __restrict__ X, const float* __restrict__ W,
                     float* __restrict__ XW) {
  const int lane = threadIdx.x & 31;
  const int wid  = (blockIdx.x * blockDim.x + threadIdx.x) >> 5;  // 0..127
  const int i0 = wid * 16;
  const int m  = lane & 15;
  const int kh = lane >> 4;           // K-half select (0: K=0,1 ; 1: K=2,3)
  v8f acc0 = {}; v8f acc1 = {};
  for (int k0 = 0; k0 < DT; k0 += 4) {
    const int ka = k0 + 2 * kh, kb = ka + 1;
    v2f a;  a.x  = X[(size_t)(i0 + m) * DT + ka];
            a.y  = X[(size_t)(i0 + m) * DT + kb];
    v2f b0; b0.x = W[ka * ET + m];      b0.y = W[kb * ET + m];
    v2f b1; b1.x = W[ka * ET + 16 + m]; b1.y = W[kb * ET + 16 + m];
    acc0 = wmma_f32_k4(a, b0, acc0);
    acc1 = wmma_f32_k4(a, b1, acc1);
  }
#pragma unroll
  for (int p = 0; p < 8; ++p) {
    const int r = i0 + p + 8 * kh;
    XW[r * ET + m]      = acc0[p];
    XW[r * ET + 16 + m] = acc1[p];
  }
}

// ------------------------------- deg[i] = sum_j A[i][j], A[i][j]=w[j*2047+i-(i>j)]
__global__ void k_deg(const float* __restrict__ w, float* __restrict__ deg) {
  const int i  = blockIdx.x * 256 + threadIdx.x;   // output row of A
  const int r0 = blockIdx.y * 64;
  float s = 0.0f;
  for (int r = r0; r < r0 + 64; ++r) {
    if (r != i) s += w[(size_t)r * NM1 + i - (i > r ? 1 : 0)];
  }
  atomicAdd(&deg[i], s);
}

// --------------------------------- dis = rsqrt(deg+1); Y = dis*XW (in place)
__global__ void k_fin_dis(float* deg_dis, float* XW) {
  const int i = blockIdx.x * blockDim.x + threadIdx.x;
  if (i >= NN) return;
  const float d = rsqrtf(deg_dis[i] + 1.0f);
  deg_dis[i] = d;
  for (int e = 0; e < ET; ++e) XW[i * ET + e] *= d;
}

// ------------- feats64[:,colOff:colOff+32] = lrelu(dis_i * (A+I)@Y + bias)
__global__ void k_gcn32(const float* __restrict__ w, const float* __restrict__ Y,
                        const float* __restrict__ dis, const float* __restrict__ bias,
                        float* __restrict__ feats64, int colOff) {
  const int lane = threadIdx.x & 31;
  const int wid  = (blockIdx.x * blockDim.x + threadIdx.x) >> 5;
  const int i0 = wid * 16;
  const int m  = lane & 15;
  const int kh = lane >> 4;
  const int i  = i0 + m;
  v8f acc0 = {}; v8f acc1 = {};
  for (int j0 = 0; j0 < NN; j0 += 4) {
    const int ja = j0 + 2 * kh, jb = ja + 1;
    v2f a;
    a.x = (i == ja) ? 1.0f : w[(size_t)ja * NM1 + i - (i > ja ? 1 : 0)];
    a.y = (i == jb) ? 1.0f : w[(size_t)jb * NM1 + i - (i > jb ? 1 : 0)];
    v2f b0, b1;
    b0.x = Y[ja * ET + m];      b0.y = Y[jb * ET + m];
    b1.x = Y[ja * ET + 16 + m]; b1.y = Y[jb * ET + 16 + m];
    acc0 = wmma_f32_k4(a, b0, acc0);
    acc1 = wmma_f32_k4(a, b1, acc1);
  }
#pragma unroll
  for (int p = 0; p < 8; ++p) {
    const int r = i0 + p + 8 * kh;
    const float dr = dis[r];
    feats64[r * 64 + colOff + m]      = lrelu(dr * acc0[p] + bias[m]);
    feats64[r * 64 + colOff + 16 + m] = lrelu(dr * acc1[p] + bias[16 + m]);
  }
}

// ---------------------------------------------------------------- one-hot ----
__global__ void k_cols(const int* __restrict__ labels, float* __restrict__ cols) {
  const int i = blockIdx.x * blockDim.x + threadIdx.x;
  if (i < NN) cols[labels[i]] = 1.0f;   // all racers store 1.0f
}

__global__ void k_hvec(const float* __restrict__ cols, const float* __restrict__ Wc,
                       float* __restrict__ hvec) {
  const int e = threadIdx.x;
  if (e < EC) {
    float s = 0.0f;
    for (int c = 0; c < NCLS; ++c) s += cols[c] * Wc[(64 + c) * EC + e];
    hvec[e] = s;
  }
}

// ----------------- Z = feats64[2048,64] @ Wc[0:64,128] + rowmask * hvec
__global__ void k_z(const float* __restrict__ feats64, const float* __restrict__ Wc,
                    const float* __restrict__ hvec, float* __restrict__ Z) {
  const int lane = threadIdx.x & 31;
  const int wid  = (blockIdx.x * blockDim.x + threadIdx.x) >> 5;
  const int i0 = wid * 16;
  const int m  = lane & 15;
  const int kh = lane >> 4;
  v8f acc[8];
#pragma unroll
  for (int t = 0; t < 8; ++t) { v8f z = {}; acc[t] = z; }
  for (int k0 = 0; k0 < 64; k0 += 4) {
    const int ka = k0 + 2 * kh, kb = ka + 1;
    v2f a; a.x = feats64[(size_t)(i0 + m) * 64 + ka];
           a.y = feats64[(size_t)(i0 + m) * 64 + kb];
#pragma unroll
    for (int t = 0; t < 8; ++t) {
      v2f b; b.x = Wc[ka * EC + t * 16 + m];
             b.y = Wc[kb * EC + t * 16 + m];
      acc[t] = wmma_f32_k4(a, b, acc[t]);
    }
  }
#pragma unroll
  for (int p = 0; p < 8; ++p) {
    const int r = i0 + p + 8 * kh;
    const float hm = (r < SUPPORT) ? 1.0f : 0.0f;
#pragma unroll
    for (int t = 0; t < 8; ++t) {
      const int c = t * 16 + m;
      Z[r * EC + c] = acc[t][p] + hm * hvec[c];
    }
  }
}

// --------------- Aw[i][j] = 1/(L1(fd_i,fd_j)+1e-5), zero diag (symmetric)
__global__ void k_dist(const float* __restrict__ feats64, float* __restrict__ Aw) {
  __shared__ float Ar[16][65];   // 65 stride -> bank-conflict free
  __shared__ float Br[16][65];
  const int tid = threadIdx.x;   // 256
  const int rbi = blockIdx.x * 16, rbj = blockIdx.y * 16;
  for (int t = tid; t < 16 * 64; t += 256) {
    const int rr = t >> 6, cc = t & 63;
    Ar[rr][cc] = feats64[(size_t)(rbi + rr) * 64 + cc];
    Br[rr][cc] = feats64[(size_t)(rbj + rr) * 64 + cc];
  }
  __syncthreads();
  const int il = tid >> 4, jl = tid & 15;
  float s = 0.0f;
#pragma unroll 8
  for (int d = 0; d < 64; ++d) s += fabsf(Ar[il][d] - Br[jl][d]);
  const int gi = rbi + il, gj = rbj + jl;
  Aw[(size_t)gi * NN + gj] = (gi == gj) ? 0.0f : 1.0f / (s + 1e-5f);
}

// -------------------------- degw[r] = rowsum(Aw), one wave per row
__global__ void k_degw(const float* __restrict__ Aw, float* __restrict__ degw) {
  const int lane = threadIdx.x & 31;
  const int row  = (blockIdx.x * blockDim.x + threadIdx.x) >> 5;
  float s = 0.0f;
  for (int j = lane; j < NN; j += 32) s += Aw[(size_t)row * NN + j];
#pragma unroll
  for (int off = 16; off > 0; off >>= 1) s += __shfl_xor(s, off, 32);
  if (lane == 0) degw[row] = s;
}

__global__ void k_fin_w(float* degw_disw, float* Z) {
  const int i = blockIdx.x * blockDim.x + threadIdx.x;
  if (i >= NN) return;
  const float d = rsqrtf(degw_disw[i] + 1.0f);
  degw_disw[i] = d;
  for (int e = 0; e < EC; ++e) Z[i * EC + e] *= d;
}

// ------------- emb = lrelu(disw_i * (Aw+I)@Zs + bc)
// 8 waves/block share an LDS-staged K-chunk of Zs (async double-buffered).
__global__ void __launch_bounds__(256)
k_gcn128(const float* __restrict__ Aw, const float* __restrict__ Zs,
         const float* __restrict__ disw, const float* __restrict__ bc,
         float* __restrict__ emb) {
  __shared__ __align__(16) float zbuf[2][CH * ZLD];
  const int tid  = threadIdx.x;
  const int lane = tid & 31;
  const int wv   = tid >> 5;                      // 0..7
  const int i0   = (blockIdx.x * 8 + wv) * 16;
  const int m    = lane & 15;
  const int kh   = lane >> 4;
  const int i    = i0 + m;

  // stage K-chunk c of Zs into buffer b: CH rows x 128 floats (b128 slots)
  auto issue = [&](int c, int b) {
    const int jc0 = c * CH;
#if USE_ASYNC_LDS
#pragma unroll
    for (int k = 0; k < 4; ++k) {
      const int idx  = tid + k * 256;             // 0..1023
      const int row  = idx >> 5;                  // 0..31
      const int slot = idx & 31;                  // 16B slots per row
      __builtin_amdgcn_global_load_async_to_lds_b128(
          (g_v4i_p)(Zs + (size_t)(jc0 + row) * EC + slot * 4),
          (l_v4i_p)(&zbuf[b][row * ZLD + slot * 4]),
          0, 0);
    }
#else
#pragma unroll
    for (int k = 0; k < 4; ++k) {
      const int idx  = tid + k * 256;
      const int row  = idx >> 5;
      const int slot = idx & 31;
      const float4 v = *(const float4*)(Zs + (size_t)(jc0 + row) * EC + slot * 4);
      *(float4*)(&zbuf[b][row * ZLD + slot * 4]) = v;
    }
#endif
  };

  v8f acc[8];
#pragma unroll
  for (int t = 0; t < 8; ++t) { v8f z = {}; acc[t] = z; }

  issue(0, 0);
  const int NC = NN / CH;                         // 64 chunks
  for (int c = 0; c < NC; ++c) {
#if USE_ASYNC_LDS
    __builtin_amdgcn_s_wait_asynccnt(0);
#endif
    __syncthreads();                              // chunk c visible to all waves
    if (c + 1 < NC) issue(c + 1, (c + 1) & 1);    // overlap next copy with WMMA
    const float* zb = &zbuf[c & 1][0];
    const int jc0 = c * CH;
    for (int jj = 0; jj < CH; jj += 4) {
      const int j0 = jc0 + jj;
      const int ja = j0 + 2 * kh, jb = ja + 1;
      v2f a;
      a.x = (i == ja) ? 1.0f : Aw[(size_t)ja * NN + i];   // Aw symmetric
      a.y = (i == jb) ? 1.0f : Aw[(size_t)jb * NN + i];
      const int ra = (jj + 2 * kh) * ZLD, rb = ra + ZLD;
#pragma unroll
      for (int t = 0; t < 8; ++t) {
        v2f bf; bf.x = zb[ra + t * 16 + m];
                bf.y = zb[rb + t * 16 + m];
        acc[t] = wmma_f32_k4(a, bf, acc[t]);
      }
    }
    // next overwrite of zbuf[c&1] happens only after the next top-of-loop
    // barrier, which every wave reaches after finishing these reads.
  }

#pragma unroll
  for (int p = 0; p < 8; ++p) {
    const int r = i0 + p + 8 * kh;
    const float dr = disw[r];
#pragma unroll
    for (int t = 0; t < 8; ++t) {
      const int cI = t * 16 + m;
      emb[r * EC + cI] = lrelu(dr * acc[t][p] + bc[cI]);
    }
  }
}

// --------------------------------- out = emb[2048,128]@Wo[128,5] + bo
__global__ void k_out(const float* __restrict__ emb, const float* __restrict__ Wo,
                      const float* __restrict__ bo, float* __restrict__ out) {
  const int t = blockIdx.x * blockDim.x + threadIdx.x;
  if (t >= NN * NCLS) return;
  const int i = t / NCLS, c = t % NCLS;
  float s = bo[c];
  for (int n = 0; n < EC; ++n) s += emb[(size_t)i * EC + n] * Wo[n * NCLS + c];
  out[t] = s;
}

// ---------------------------------------------------------------- launch ----
extern "C" void kernel_launch(void* const* d_in, const int* in_sizes, int n_in,
                              void* d_out, int out_size, void* d_ws, size_t ws_size,
                              hipStream_t stream) {
  const float* tfeat = (const float*)d_in[0];
  // d_in[1] = edge_index (unused: full graph structure is analytic)
  const float* wt    = (const float*)d_in[2];
  const float* ffeat = (const float*)d_in[3];
  const float* wf    = (const float*)d_in[4];
  const int*   labels= (const int*)  d_in[5];
  const float* Wt    = (const float*)d_in[8];
  const float* bt    = (const float*)d_in[9];
  const float* Wf    = (const float*)d_in[10];
  const float* bf    = (const float*)d_in[11];
  const float* Wc    = (const float*)d_in[12];
  const float* bc    = (const float*)d_in[13];
  const float* Wo    = (const float*)d_in[14];
  const float* bo    = (const float*)d_in[15];
  float* out = (float*)d_out;
  float* ws  = (float*)d_ws;

  // workspace layout (floats)
  constexpr size_t OFF_F64  = 0;                          // feats64 [2048,64]
  constexpr size_t OFF_XWT  = OFF_F64 + (size_t)NN * 64;  // XW_t -> Y_t
  constexpr size_t OFF_XWF  = OFF_XWT + (size_t)NN * ET;  // XW_f -> Y_f
  constexpr size_t OFF_DIST = OFF_XWF + (size_t)NN * ET;  // deg_t -> dis_t
  constexpr size_t OFF_DISF = OFF_DIST + NN;              // deg_f -> dis_f
  constexpr size_t OFF_COLS = OFF_DISF + NN;              // 8
  constexpr size_t OFF_HVEC = OFF_COLS + 8;               // 128
  constexpr size_t OFF_Z    = OFF_HVEC + EC;              // Z -> Zs [2048,128]
  constexpr size_t OFF_DW   = OFF_Z + (size_t)NN * EC;    // degw -> disw
  constexpr size_t OFF_EMB  = OFF_DW + NN;                // emb [2048,128]
  constexpr size_t OFF_AW   = OFF_EMB + (size_t)NN * EC;  // Aw [2048,2048]

  float* f64   = ws + OFF_F64;
  float* xwt   = ws + OFF_XWT;
  float* xwf   = ws + OFF_XWF;
  float* dist_ = ws + OFF_DIST;
  float* disf_ = ws + OFF_DISF;
  float* cols  = ws + OFF_COLS;
  float* hvec  = ws + OFF_HVEC;
  float* Z     = ws + OFF_Z;
  float* dw    = ws + OFF_DW;
  float* emb   = ws + OFF_EMB;
  float* Aw    = ws + OFF_AW;

  hipLaunchKernelGGL(k_init, dim3(8), dim3(256), 0, stream, dist_, disf_, dw, cols);

  // feature projections (WMMA f32)
  hipLaunchKernelGGL(k_xw, dim3(32), dim3(128), 0, stream, tfeat, Wt, xwt);
  hipLaunchKernelGGL(k_xw, dim3(32), dim3(128), 0, stream, ffeat, Wf, xwf);

  // degrees of dense adjacencies (coalesced partial sums)
  hipLaunchKernelGGL(k_deg, dim3(8, 32), dim3(256), 0, stream, wt, dist_);
  hipLaunchKernelGGL(k_deg, dim3(8, 32), dim3(256), 0, stream, wf, disf_);
  hipLaunchKernelGGL(k_fin_dis, dim3(8), dim3(256), 0, stream, dist_, xwt);
  hipLaunchKernelGGL(k_fin_dis, dim3(8), dim3(256), 0, stream, disf_, xwf);

  // GCN aggregations -> feats64 = [te | fe]
  hipLaunchKernelGGL(k_gcn32, dim3(32), dim3(128), 0, stream, wt, xwt, dist_, bt, f64, 0);
  hipLaunchKernelGGL(k_gcn32, dim3(32), dim3(128), 0, stream, wf, xwf, disf_, bf, f64, 32);

  // one-hot contribution vector
  hipLaunchKernelGGL(k_cols, dim3(8), dim3(256), 0, stream, labels, cols);
  hipLaunchKernelGGL(k_hvec, dim3(1), dim3(128), 0, stream, cols, Wc, hvec);

  // Z = feats @ Wc (with implicit one-hot block)
  hipLaunchKernelGGL(k_z, dim3(32), dim3(128), 0, stream, f64, Wc, hvec, Z);

  // pairwise L1 -> Aw; its degrees; scale Z
  hipLaunchKernelGGL(k_dist, dim3(128, 128), dim3(256), 0, stream, f64, Aw);
  hipLaunchKernelGGL(k_degw, dim3(256), dim3(256), 0, stream, Aw, dw);
  hipLaunchKernelGGL(k_fin_w, dim3(8), dim3(256), 0, stream, dw, Z);

  // main aggregation (WMMA f32, async LDS-staged B) + head
  hipLaunchKernelGGL(k_gcn128, dim3(16), dim3(256), 0, stream, Aw, Z, dw, bc, emb);
  hipLaunchKernelGGL(k_out, dim3(40), dim3(256), 0, stream, emb, Wo, bo, out);
}